// VoxelGenerator_65128884076580
// MI455X (gfx1250) — compile-verified
//
#include <hip/hip_runtime.h>
#include <hip/hip_bf16.h>
#include <stdint.h>

#define GRID_X 432
#define GRID_Y 496
#define GRID_Z 1
#define NUM_VOX (GRID_X * GRID_Y * GRID_Z)   // 214272
#define MAX_PTS 32
#define MAX_VOX 40000
#define CHUNK 1024
#define HASH_SZ 4096
#define FLAG_CAP 512

typedef unsigned int u32x4 __attribute__((ext_vector_type(4)));
typedef unsigned int u32x8 __attribute__((ext_vector_type(8)));

// ---------------------------------------------------------------------------
// helpers
// ---------------------------------------------------------------------------
__device__ __forceinline__ bool coord_of(float px, float py, float pz,
                                         int& cx, int& cy, int& cz) {
  cx = (int)floorf((px - 0.0f)     / 0.16f);
  cy = (int)floorf((py + 39.68f)   / 0.16f);
  cz = (int)floorf((pz + 3.0f)     / 4.0f);
  return (cx >= 0) && (cx < GRID_X) &&
         (cy >= 0) && (cy < GRID_Y) &&
         (cz >= 0) && (cz < GRID_Z);
}

// CDNA5 Tensor Data Mover: one DMA descriptor moves a contiguous run of
// `remain8`-bounded 8-byte elements (tile = 2048 x 8B = 16KB) into LDS.
// Tracked by TENSORcnt. OOB-vs-tensor_dim0 reads return zero (safe tail).
__device__ __forceinline__ void tdm_load_chunk(unsigned int lds_addr,
                                               unsigned long long gaddr,
                                               unsigned int remain8) {
  u32x4 g0;
  g0[0] = 1u;                                   // count=1 valid descriptor
  g0[1] = lds_addr;                             // LDS destination (bytes)
  g0[2] = (unsigned int)gaddr;                  // global_addr[31:0]
  g0[3] = ((unsigned int)(gaddr >> 32) & 0x01ffffffu) | 0x80000000u; // [56:32] | type=2
  u32x8 g1;
  g1[0] = 3u << 16;                             // wg_mask=0, data_size=3 (8B)
  g1[1] = (remain8 & 0xffffu) << 16;            // tensor_dim0[15:0]
  g1[2] = (remain8 >> 16) | (1u << 16);         // tensor_dim0[31:16], tensor_dim1=1
  g1[3] = ((unsigned int)(CHUNK * 2)) << 16;    // tile_dim0 = 2048 elements
  g1[4] = 1u;                                   // tile_dim1=1, tile_dim2=0
  g1[5] = 0u;                                   // tensor_dim0_stride lo
  g1[6] = 0u;                                   // stride hi / dim1_stride lo
  g1[7] = 0u;                                   // dim1_stride hi
  asm volatile("tensor_load_to_lds %0, %1" :: "s"(g0), "s"(g1) : "memory");
}

// ---------------------------------------------------------------------------
// 1) output init: voxels=0, coords=-1, valid=0, npts=0
// ---------------------------------------------------------------------------
__global__ void vg_init_out(float* __restrict__ out, long long total,
                            long long cbeg, long long cend) {
  long long i = (long long)blockIdx.x * blockDim.x + threadIdx.x;
  long long stride = (long long)gridDim.x * blockDim.x;
  for (; i < total; i += stride)
    out[i] = (i >= cbeg && i < cend) ? -1.0f : 0.0f;
}

// ---------------------------------------------------------------------------
// 2) workspace init
// ---------------------------------------------------------------------------
__global__ void vg_init_ws(int* __restrict__ first_idx, int* __restrict__ slot,
                           int* __restrict__ pcount, int n) {
  int i = blockIdx.x * blockDim.x + threadIdx.x;
  if (i < n) {
    first_idx[i] = 0x7fffffff;
    slot[i] = -1;
    pcount[i] = 0;
  }
}

// ---------------------------------------------------------------------------
// 3) pass1: per-voxel min point index (order-free atomics)
// ---------------------------------------------------------------------------
__global__ void vg_pass1(const float4* __restrict__ pts,
                         const int* __restrict__ validIn,
                         int* __restrict__ first_idx, int B_, int N_) {
  int gid = blockIdx.x * blockDim.x + threadIdx.x;
  if (gid >= B_ * N_) return;
  int b = gid / N_;
  int i = gid - b * N_;
  if (i >= validIn[b]) return;
  float4 p = pts[gid];
  int cx, cy, cz;
  if (!coord_of(p.x, p.y, p.z, cx, cy, cz)) return;
  int lin = (cz * GRID_Y + cy) * GRID_X + cx;
  atomicMin(first_idx + (size_t)b * NUM_VOX + lin, i);
}

// ---------------------------------------------------------------------------
// 4) sequential-order scan: one block per batch, 1024 threads.
//    Point chunks are staged into LDS by the Tensor Data Mover (one
//    tensor_load_to_lds per chunk, double-buffered, issued by wave 0 only).
// ---------------------------------------------------------------------------
__global__ void __launch_bounds__(CHUNK) vg_scan(
    const float4* __restrict__ pts, const int* __restrict__ validIn,
    const int* __restrict__ first_idx, int* __restrict__ slot,
    int* __restrict__ pcount, unsigned char* __restrict__ pslot,
    float* __restrict__ coords_out, float* __restrict__ valid_out,
    int N_, int totalPts) {
  __shared__ float4 spts[2][CHUNK];                  // 32 KB double buffer
  __shared__ int slin[CHUNK];                        // 4 KB
  __shared__ unsigned int shash[HASH_SZ];            // 16 KB
  __shared__ unsigned long long sflag[FLAG_CAP];     // 4 KB (lin<<32 | tid)
  __shared__ int sflagcnt;
  __shared__ int swavetot[32];
  __shared__ int swaveoff[32];
  __shared__ int schunktot;
  __shared__ int scarry;

  const int b = blockIdx.x;
  const int tid = threadIdx.x;
  const int lane = tid & 31;
  const int wav = tid >> 5;
  const int nvalid = validIn[b];

  const int* fidx = first_idx + (size_t)b * NUM_VOX;
  int* slt = slot + (size_t)b * NUM_VOX;
  int* pcnt = pcount + (size_t)b * NUM_VOX;
  unsigned char* psl = pslot + (size_t)b * N_;
  float* cout = coords_out + (size_t)b * MAX_VOX * 3;

  if (tid == 0) scarry = 0;

  const unsigned long long gbase = (unsigned long long)(size_t)pts;
  const unsigned int ldsBase = (unsigned int)(size_t)&spts[0][0];
  const int nChunks = (N_ + CHUNK - 1) / CHUNK;
  // scalar (wave-uniform) leader test -> guaranteed s_cbranch skip for waves>0
  const bool leader = (__builtin_amdgcn_readfirstlane(wav) == 0);

  // prologue: TDM-stage chunk 0 into buffer 0
  if (leader) {
    int e0 = (b * N_) * 2;  // 8B-element index of chunk start in whole buffer
    tdm_load_chunk(ldsBase, gbase + (unsigned long long)e0 * 8ull,
                   (unsigned int)(totalPts * 2 - e0));
  }

  for (int c = 0; c < nChunks; ++c) {
    const int buf = c & 1;
    if (leader) {
      if (c + 1 < nChunks) {   // prefetch next chunk into the other buffer
        int e = (b * N_ + (c + 1) * CHUNK) * 2;
        tdm_load_chunk(ldsBase + (unsigned int)((buf ^ 1) * CHUNK * 16),
                       gbase + (unsigned long long)e * 8ull,
                       (unsigned int)(totalPts * 2 - e));
        __builtin_amdgcn_s_wait_tensorcnt(1);  // current chunk landed
      } else {
        __builtin_amdgcn_s_wait_tensorcnt(0);
      }
    }
    __syncthreads();  // (A) chunk data visible; safe reuse of slin/shash/sflag

    // phase 1: hash clear + coords + slin
    for (int h = tid; h < HASH_SZ; h += CHUNK) shash[h] = 0u;
    if (tid == 0) sflagcnt = 0;

    const int i = c * CHUNK + tid;  // point index within batch
    float4 p = spts[buf][tid];
    int cx, cy, cz;
    bool inr = (i < N_) && (i < nvalid) && coord_of(p.x, p.y, p.z, cx, cy, cz);
    int lin = inr ? ((cz * GRID_Y + cy) * GRID_X + cx) : -1;
    slin[tid] = lin;
    __syncthreads();  // (B)

    unsigned int hb = 0;
    if (inr) {
      hb = ((unsigned int)lin * 2654435761u) >> 20;  // 12-bit bucket
      atomicAdd(&shash[hb], 1u);
    }
    __syncthreads();  // (C)

    bool flagged = inr && (shash[hb] > 1u);
    if (flagged) {
      int fi = atomicAdd(&sflagcnt, 1);
      if (fi < FLAG_CAP)
        sflag[fi] = ((unsigned long long)(unsigned int)lin << 32) |
                    (unsigned long long)(unsigned int)tid;
    }
    __syncthreads();  // (D)

    int cntBefore = 0, cntTotal = 1;
    bool isLast = true;
    const int fcnt = sflagcnt;
    if (flagged) {
      if (fcnt <= FLAG_CAP) {
        for (int j = 0; j < fcnt; ++j) {
          unsigned long long e = sflag[j];
          if ((int)(e >> 32) == lin) {
            int jt = (int)(unsigned int)(e & 0xffffffffu);
            if (jt < tid) cntBefore++;
            else if (jt > tid) { isLast = false; }
            if (jt != tid) cntTotal++;
          }
        }
      } else {  // extremely rare overflow: exact fallback over whole chunk
        for (int j = 0; j < CHUNK; ++j) {
          if (slin[j] == lin) {
            if (j < tid) cntBefore++;
            else if (j > tid) { isLast = false; }
            if (j != tid) cntTotal++;
          }
        }
      }
    }

    // per-point slot within voxel = carried count + within-chunk ordinal
    int baseCnt = 0;
    if (inr)
      baseCnt = __hip_atomic_load(&pcnt[lin], __ATOMIC_RELAXED,
                                  __HIP_MEMORY_SCOPE_AGENT);
    if (i < N_) {
      int ps = inr ? (baseCnt + cntBefore) : 0;
      psl[i] = (unsigned char)(ps > 255 ? 255 : ps);
    }
    __syncthreads();  // (E) all base reads precede updates
    if (inr && isLast)
      __hip_atomic_fetch_add(&pcnt[lin], cntTotal, __ATOMIC_RELAXED,
                             __HIP_MEMORY_SCOPE_AGENT);

    // voxel rank = exclusive prefix of is_first in point-index order
    bool isFirst = inr && (fidx[lin] == i);
    unsigned long long bal = __ballot((int)isFirst);
    int lanePre = __popcll(bal & ((1ull << lane) - 1ull));
    if (lane == 0) swavetot[wav] = __popcll(bal);
    __syncthreads();  // (F)
    if (tid == 0) {
      int acc = 0;
      for (int w = 0; w < 32; ++w) { swaveoff[w] = acc; acc += swavetot[w]; }
      schunktot = acc;
    }
    __syncthreads();  // (G)

    int rank = scarry + swaveoff[wav] + lanePre;
    if (isFirst && rank < MAX_VOX) {
      slt[lin] = rank;
      int rem = lin % (GRID_X * GRID_Y);
      cout[(size_t)rank * 3 + 0] = (float)(lin / (GRID_X * GRID_Y));  // z
      cout[(size_t)rank * 3 + 1] = (float)(rem / GRID_X);             // y
      cout[(size_t)rank * 3 + 2] = (float)(rem % GRID_X);             // x
    }
    __syncthreads();  // (H) everyone read scarry before update
    if (tid == 0) scarry += schunktot;
  }

  __syncthreads();
  if (tid == 0) {
    int v = scarry;
    if (v > MAX_VOX) v = MAX_VOX;
    valid_out[b] = (float)v;
  }
}

// ---------------------------------------------------------------------------
// 5) scatter kept points into voxels output (16B vector stores)
// ---------------------------------------------------------------------------
__global__ void vg_scatter(const float4* __restrict__ pts,
                           const int* __restrict__ validIn,
                           const int* __restrict__ slot,
                           const unsigned char* __restrict__ pslot,
                           float* __restrict__ voxels, int B_, int N_) {
  int gid = blockIdx.x * blockDim.x + threadIdx.x;
  if (gid >= B_ * N_) return;
  int b = gid / N_;
  int i = gid - b * N_;
  if (i >= validIn[b]) return;
  float4 p = pts[gid];
  int cx, cy, cz;
  if (!coord_of(p.x, p.y, p.z, cx, cy, cz)) return;
  int lin = (cz * GRID_Y + cy) * GRID_X + cx;
  int v = slot[(size_t)b * NUM_VOX + lin];
  if (v < 0) return;  // unranked or overflow voxel
  int ps = pslot[gid];
  if (ps >= MAX_PTS) return;
  float4* dst = (float4*)(voxels +
      (((size_t)b * MAX_VOX + (size_t)v) * MAX_PTS + (size_t)ps) * 4);
  *dst = p;
}

// ---------------------------------------------------------------------------
// 6) per-voxel point counts
// ---------------------------------------------------------------------------
__global__ void vg_npts(const int* __restrict__ slot,
                        const int* __restrict__ pcount,
                        float* __restrict__ npts_out, int B_) {
  int gid = blockIdx.x * blockDim.x + threadIdx.x;
  if (gid >= B_ * NUM_VOX) return;
  int s = slot[gid];
  if (s < 0) return;
  int b = gid / NUM_VOX;
  int c = pcount[gid];
  if (c > MAX_PTS) c = MAX_PTS;
  npts_out[(size_t)b * MAX_VOX + s] = (float)c;
}

// ---------------------------------------------------------------------------
// host entry
// ---------------------------------------------------------------------------
extern "C" void kernel_launch(void* const* d_in, const int* in_sizes, int n_in,
                              void* d_out, int out_size, void* d_ws,
                              size_t ws_size, hipStream_t stream) {
  const float* points = (const float*)d_in[0];
  const int* validIn = (const int*)d_in[1];
  const int F_ = 4;
  const int B_ = in_sizes[1];
  const int totalPts = in_sizes[0] / F_;
  const int N_ = totalPts / B_;

  // workspace layout
  const int nvoxtot = B_ * NUM_VOX;
  int* first_idx = (int*)d_ws;
  int* slot = first_idx + nvoxtot;
  int* pcount = slot + nvoxtot;
  unsigned char* pslot = (unsigned char*)(pcount + nvoxtot);

  // output layout (flat, return order; int outputs stored as fp32 values)
  float* out = (float*)d_out;
  const long long voxN = (long long)B_ * MAX_VOX * MAX_PTS * F_;
  const long long coordsN = (long long)B_ * MAX_VOX * 3;
  const long long nptsN = (long long)B_ * MAX_VOX;
  float* coords_out = out + voxN;
  float* valid_out = out + voxN + coordsN;
  float* npts_out = valid_out + B_;
  const long long totalOut = voxN + coordsN + B_ + nptsN;

  vg_init_out<<<4096, 256, 0, stream>>>(out, totalOut, voxN, voxN + coordsN);
  vg_init_ws<<<(nvoxtot + 255) / 256, 256, 0, stream>>>(first_idx, slot,
                                                        pcount, nvoxtot);
  vg_pass1<<<(totalPts + 255) / 256, 256, 0, stream>>>(
      (const float4*)points, validIn, first_idx, B_, N_);
  vg_scan<<<B_, CHUNK, 0, stream>>>((const float4*)points, validIn, first_idx,
                                    slot, pcount, pslot, coords_out, valid_out,
                                    N_, totalPts);
  vg_scatter<<<(totalPts + 255) / 256, 256, 0, stream>>>(
      (const float4*)points, validIn, slot, pslot, out, B_, N_);
  vg_npts<<<(nvoxtot + 255) / 256, 256, 0, stream>>>(slot, pcount, npts_out,
                                                     B_);
}